// GCNModel_NoEdges_25366076850805
// MI455X (gfx1250) — compile-verified
//
#include <hip/hip_runtime.h>
#include <hip/hip_bf16.h>

typedef float v2f __attribute__((ext_vector_type(2)));
typedef float v8f __attribute__((ext_vector_type(8)));

#define NEG_SLOPE 0.01f
#define BN_EPS 1e-5f

// ---------------------------------------------------------------------------
// Degree / normalization
// ---------------------------------------------------------------------------
__global__ void k_init_deg(float* __restrict__ deg, int N) {
  int i = blockIdx.x * blockDim.x + threadIdx.x;
  if (i < N) deg[i] = 1.0f;  // self-loop contributes 1 to every node's degree
}

__global__ void k_count_deg(const int* __restrict__ dst, float* __restrict__ deg, int E) {
  int i = blockIdx.x * blockDim.x + threadIdx.x;
  if (i < E) atomicAdd(&deg[dst[i]], 1.0f);
}

__global__ void k_rsqrt_inplace(float* __restrict__ deg, int N) {
  int i = blockIdx.x * blockDim.x + threadIdx.x;
  if (i < N) deg[i] = rsqrtf(deg[i]);  // deg -> dinv in place
}

// ---------------------------------------------------------------------------
// FP32 WMMA GEMM, strip-mined: one wave32 computes a 16 x (NT*16) output strip.
// C[M,N] = A[M,K] @ B[K,N] (+bias) (+LeakyReLU),  N = NT*16, K compile-time.
//
// Per K-step of 4 the wave loads ONE A fragment and feeds NT back-to-back
// V_WMMA_F32_16X16X4_F32 ops (8x fewer A fetches than one-wave-per-tile,
// much higher wmma:vmem density).
//
// Fragment layout (CDNA5 ISA 7.12.2, 32-bit, wave32):
//   A 16x4 : lanes 0-15 -> M=lane, {v.x,v.y}={K0,K1}; lanes 16-31 -> {K2,K3}
//   B 4x16 : lanes 0-15 -> N=lane, {v.x,v.y}={K0,K1}; lanes 16-31 -> {K2,K3}
//   C 16x16: VGPR r, lanes 0-15 -> M=r,  N=lane; lanes 16-31 -> M=r+8, N=lane-16
// ---------------------------------------------------------------------------
template <int K, int NT, int ACT>  // ACT: 0 = none, 1 = LeakyReLU
__global__ void k_wmma_gemm_strip(const float* __restrict__ A,
                                  const float* __restrict__ B,
                                  const float* __restrict__ bias,
                                  float* __restrict__ C, int M) {
  constexpr int N = NT * 16;
  const int lane  = threadIdx.x & 31;
  const int tileM = (blockIdx.x * blockDim.x + threadIdx.x) >> 5;  // wave id
  if (tileM * 16 >= M) return;  // wave-uniform -> EXEC stays all-1s for WMMA

  const int half = lane >> 4;
  const int lq   = lane & 15;
  const int row  = tileM * 16 + lq;  // A row handled by this lane

  const float* __restrict__ Arow = A + (size_t)row * K;

  v8f acc[NT];
#pragma unroll
  for (int t = 0; t < NT; ++t) acc[t] = (v8f){0.f, 0.f, 0.f, 0.f, 0.f, 0.f, 0.f, 0.f};

#pragma unroll
  for (int k0 = 0; k0 < K; k0 += 4) {
    const int kb = k0 + half * 2;
    v2f a;
    a.x = Arow[kb];
    a.y = Arow[kb + 1];
    const float* __restrict__ B0 = B + (size_t)kb * N + lq;
    const float* __restrict__ B1 = B + (size_t)(kb + 1) * N + lq;
#pragma unroll
    for (int t = 0; t < NT; ++t) {
      v2f b;
      b.x = B0[t * 16];
      b.y = B1[t * 16];
      acc[t] = __builtin_amdgcn_wmma_f32_16x16x4_f32(false, a, false, b,
                                                     (short)0, acc[t], false, false);
    }
  }

#pragma unroll
  for (int t = 0; t < NT; ++t) {
    const int col = t * 16 + lq;
    const float bv = bias ? bias[col] : 0.f;
#pragma unroll
    for (int v = 0; v < 8; ++v) {
      const int r = tileM * 16 + v + half * 8;
      float val = acc[t][v] + bv;
      if (ACT == 1) val = val > 0.f ? val : NEG_SLOPE * val;
      C[(size_t)r * N + col] = val;
    }
  }
}

// ---------------------------------------------------------------------------
// Self-loop init: agg[n,:] = dinv[n]^2 * m[n,:]   (H = 128, float4 per lane)
// ---------------------------------------------------------------------------
__global__ void k_selfloop(const float* __restrict__ dinv,
                           const float* __restrict__ m,
                           float* __restrict__ agg, int N) {
  int t = blockIdx.x * blockDim.x + threadIdx.x;
  int n = t >> 5;
  if (n >= N) return;
  int q = t & 31;  // 32 float4 chunks cover H=128
  float w = dinv[n] * dinv[n];
  float4 v = ((const float4*)m)[(size_t)n * 32 + q];
  ((float4*)agg)[(size_t)n * 32 + q] =
      make_float4(w * v.x, w * v.y, w * v.z, w * v.w);
}

// ---------------------------------------------------------------------------
// Edge gather-weight-scatter: agg[d,:] += dinv[s]*dinv[d] * m[s,:]
// One wave per edge; lane handles 4 consecutive features (H = 128).
// ---------------------------------------------------------------------------
__global__ void k_edge_scatter(const int* __restrict__ src,
                               const int* __restrict__ dst,
                               const float* __restrict__ dinv,
                               const float* __restrict__ m,
                               float* __restrict__ agg, int E) {
  int t = blockIdx.x * blockDim.x + threadIdx.x;
  int e = t >> 5;
  if (e >= E) return;
  int lane = t & 31;
  int s = src[e];
  int d = dst[e];
  float w = dinv[s] * dinv[d];
  float4 v = ((const float4*)m)[(size_t)s * 32 + lane];
  float* ag = agg + (size_t)d * 128 + lane * 4;
  atomicAdd(ag + 0, w * v.x);
  atomicAdd(ag + 1, w * v.y);
  atomicAdd(ag + 2, w * v.z);
  atomicAdd(ag + 3, w * v.w);
}

// ---------------------------------------------------------------------------
// BatchNorm (eval) + LeakyReLU:  h = lrelu((agg + convB - mean)*rsqrt(var+eps)*gamma + beta)
// ---------------------------------------------------------------------------
__global__ void k_bn_lrelu(const float* __restrict__ agg,
                           const float* __restrict__ convB,
                           const float* __restrict__ gamma,
                           const float* __restrict__ beta,
                           const float* __restrict__ mean,
                           const float* __restrict__ var,
                           float* __restrict__ h, int total) {
  int i = blockIdx.x * blockDim.x + threadIdx.x;
  if (i >= total) return;
  int f = i & 127;  // H = 128
  float a = agg[i] + convB[f];
  float x = (a - mean[f]) * rsqrtf(var[f] + BN_EPS) * gamma[f] + beta[f];
  h[i] = x > 0.f ? x : NEG_SLOPE * x;
}

// ---------------------------------------------------------------------------
// Global add pool
// ---------------------------------------------------------------------------
__global__ void k_zero(float* __restrict__ p, int n) {
  int i = blockIdx.x * blockDim.x + threadIdx.x;
  if (i < n) p[i] = 0.f;
}

__global__ void k_pool(const float* __restrict__ h, const int* __restrict__ batch,
                       float* __restrict__ g, int N) {
  int t = blockIdx.x * blockDim.x + threadIdx.x;
  int n = t >> 5;
  if (n >= N) return;
  int lane = t & 31;
  int b = batch[n];
  float4 v = ((const float4*)h)[(size_t)n * 32 + lane];
  float* gp = g + (size_t)b * 128 + lane * 4;
  atomicAdd(gp + 0, v.x);
  atomicAdd(gp + 1, v.y);
  atomicAdd(gp + 2, v.z);
  atomicAdd(gp + 3, v.w);
}

// ---------------------------------------------------------------------------
// Final tiny GEMM: out[G,OUT] = z1[G,H2] @ W2[H2,OUT] + b2  (G=64, H2=64, OUT=8)
// ---------------------------------------------------------------------------
__global__ void k_final_mlp(const float* __restrict__ z1,
                            const float* __restrict__ W2,
                            const float* __restrict__ b2,
                            float* __restrict__ out, int G, int H2, int OUT) {
  int t = blockIdx.x * blockDim.x + threadIdx.x;
  if (t >= G * OUT) return;
  int gi = t / OUT, o = t % OUT;
  float acc = b2[o];
  for (int k = 0; k < H2; ++k) acc += z1[(size_t)gi * H2 + k] * W2[(size_t)k * OUT + o];
  out[(size_t)gi * OUT + o] = acc;
}

// ---------------------------------------------------------------------------
// Host launch
// ---------------------------------------------------------------------------
static inline size_t align_up(size_t v) { return (v + 255) & ~(size_t)255; }

extern "C" void kernel_launch(void* const* d_in, const int* in_sizes, int n_in,
                              void* d_out, int out_size, void* d_ws, size_t ws_size,
                              hipStream_t stream) {
  const float* x       = (const float*)d_in[0];
  const int*   ei      = (const int*)d_in[1];
  const int*   batch   = (const int*)d_in[2];
  const float* W_emb   = (const float*)d_in[3];
  const float* b_emb   = (const float*)d_in[4];
  const float* convW   = (const float*)d_in[5];
  const float* convB   = (const float*)d_in[6];
  const float* bn_gamma= (const float*)d_in[7];
  const float* bn_beta = (const float*)d_in[8];
  const float* bn_mean = (const float*)d_in[9];
  const float* bn_var  = (const float*)d_in[10];
  const float* W1      = (const float*)d_in[11];
  const float* b1      = (const float*)d_in[12];
  const float* W2      = (const float*)d_in[13];
  const float* b2      = (const float*)d_in[14];
  float* out = (float*)d_out;

  const int N    = in_sizes[2];            // 50000
  const int E    = in_sizes[1] / 2;        // 800000
  const int H    = in_sizes[4];            // 128 (compile-time in GEMMs)
  const int L    = in_sizes[6] / H;        // 3
  const int H2   = in_sizes[12];           // 64
  const int OUT  = in_sizes[14];           // 8
  const int G    = out_size / OUT;         // 64

  const int* src = ei;       // edge_index[0]
  const int* dst = ei + E;   // edge_index[1]

  // Workspace carve-out
  char* ws = (char*)d_ws;
  float* dinv = (float*)ws; ws += align_up((size_t)N * 4);
  float* h    = (float*)ws; ws += align_up((size_t)N * H * 4);
  float* m    = (float*)ws; ws += align_up((size_t)N * H * 4);
  float* agg  = (float*)ws; ws += align_up((size_t)N * H * 4);
  float* g    = (float*)ws; ws += align_up((size_t)G * H * 4);
  float* z1   = (float*)ws; ws += align_up((size_t)G * H2 * 4);

  const int BLK = 256;
  const int wavesPerBlk = BLK / 32;
  const int rowTiles = (N + 15) / 16;                       // 3125
  const int gemmBlocks = (rowTiles + wavesPerBlk - 1) / wavesPerBlk;

  // 1) degree -> dinv
  k_init_deg<<<(N + BLK - 1) / BLK, BLK, 0, stream>>>(dinv, N);
  k_count_deg<<<(E + BLK - 1) / BLK, BLK, 0, stream>>>(dst, dinv, E);
  k_rsqrt_inplace<<<(N + BLK - 1) / BLK, BLK, 0, stream>>>(dinv, N);

  // 2) embedding: h = x @ W_emb + b_emb   [N,64]@[64,128]  (WMMA f32, 16x128 strips)
  k_wmma_gemm_strip<64, 8, 0><<<gemmBlocks, BLK, 0, stream>>>(x, W_emb, b_emb, h, N);

  // 3) conv layers
  for (int i = 0; i < L; ++i) {
    // m = h @ convW[i]   [N,128]@[128,128]
    k_wmma_gemm_strip<128, 8, 0><<<gemmBlocks, BLK, 0, stream>>>(
        h, convW + (size_t)i * H * H, nullptr, m, N);

    k_selfloop<<<((size_t)N * 32 + BLK - 1) / BLK, BLK, 0, stream>>>(dinv, m, agg, N);

    k_edge_scatter<<<((size_t)E * 32 + BLK - 1) / BLK, BLK, 0, stream>>>(
        src, dst, dinv, m, agg, E);

    k_bn_lrelu<<<((size_t)N * H + BLK - 1) / BLK, BLK, 0, stream>>>(
        agg, convB + (size_t)i * H, bn_gamma + (size_t)i * H, bn_beta + (size_t)i * H,
        bn_mean + (size_t)i * H, bn_var + (size_t)i * H, h, N * H);
  }

  // 4) global add pool
  k_zero<<<(G * H + BLK - 1) / BLK, BLK, 0, stream>>>(g, G * H);
  k_pool<<<((size_t)N * 32 + BLK - 1) / BLK, BLK, 0, stream>>>(h, batch, g, N);

  // 5) MLP head: z1 = lrelu(g @ W1 + b1)  [64,128]@[128,64]  (WMMA f32, 16x64 strips)
  {
    int tilesG = (G + 15) / 16;  // 4 waves
    k_wmma_gemm_strip<128, 4, 1><<<(tilesG + wavesPerBlk - 1) / wavesPerBlk, BLK, 0, stream>>>(
        g, W1, b1, z1, G);
  }
  k_final_mlp<<<(G * OUT + BLK - 1) / BLK, BLK, 0, stream>>>(z1, W2, b2, out, G, H2, OUT);
}